// NonMaxSuppression_11914239279656
// MI455X (gfx1250) — compile-verified
//
#include <hip/hip_runtime.h>
#include <stdint.h>

// Image size (fixed by the reference)
constexpr int IMG_H = 4096;
constexpr int IMG_W = 4096;

// Tile config: 512 threads (16 wave32s), each thread -> 4 pixels (float4)
constexpr int TILE_X   = 128;                 // output tile width (pixels)
constexpr int TILE_Y   = 16;                  // output tile height (rows)
constexpr int HALO     = 4;                   // halo floats each side (16B aligned)
constexpr int LDS_W    = TILE_X + 2 * HALO;   // 136 floats per staged row
constexpr int LDS_P    = LDS_W;               // pitch == row length (TDM-contiguous)
constexpr int LDS_ROWS = TILE_Y + 2;          // 18 rows (1 halo row each side)
constexpr int NVEC     = LDS_W / 4;           // 34 b128 loads per row
constexpr int NLOADS   = NVEC * LDS_ROWS;     // 612 async loads per edge tile
constexpr int NTHREADS = 512;

typedef __attribute__((ext_vector_type(4))) float    v4f;
typedef __attribute__((ext_vector_type(4))) uint32_t su4;
typedef __attribute__((ext_vector_type(8))) uint32_t su8;

__device__ __forceinline__ int clampi(int v, int lo, int hi) {
    return v < lo ? lo : (v > hi ? hi : v);
}

__global__ __launch_bounds__(NTHREADS)
void nms_kernel(const float* __restrict__ rel,
                const float* __restrict__ rep,
                const float* __restrict__ rel_thr_p,
                const float* __restrict__ rep_thr_p,
                float* __restrict__ out)
{
    __shared__ float smem[LDS_ROWS * LDS_P];

    const int gx     = blockIdx.x * TILE_X;
    const int gy     = blockIdx.y * TILE_Y;
    const int tid    = (int)threadIdx.y * 32 + (int)threadIdx.x;
    const int base_x = gx - HALO;             // global column of LDS column 0

    const int px = gx + (int)threadIdx.x * 4; // first of this thread's 4 pixels
    const int py = gy + (int)threadIdx.y;

    // Issue independent streaming loads before staging so they overlap the DMA.
    const v4f   r4      = __builtin_nontemporal_load(
                              (const v4f*)(rel + (size_t)py * IMG_W + px));
    const float rel_thr = rel_thr_p[0];
    const float rep_thr = rep_thr_p[0];

    const bool interior =
        (blockIdx.x > 0) && (blockIdx.x < gridDim.x - 1) &&
        (blockIdx.y > 0) && (blockIdx.y < gridDim.y - 1);

    if (interior) {
        // ---- One Tensor Data Mover descriptor stages the whole 136x18 tile.
        // Fully in-bounds (no clamping needed), rows written contiguously to LDS.
        if (threadIdx.y == 0) {               // exactly one wave32 programs TDM
            const uint64_t gaddr =
                (uint64_t)(uintptr_t)(rep + (size_t)(gy - 1) * IMG_W + base_x);
            const uint32_t lds0 = (uint32_t)(uintptr_t)(&smem[0]);

            // D# group 0: count=1 | lds_addr | global_addr[56:0] | type=2
            su4 g0;
            g0[0] = 1u;                                    // count=1, user mode
            g0[1] = lds0;                                  // LDS byte address
            g0[2] = (uint32_t)gaddr;                       // global addr [31:0]
            g0[3] = ((uint32_t)(gaddr >> 32) & 0x01FFFFFFu) // global addr [56:32]
                    | 0x80000000u;                          // type=2 ("image")
            // D# group 1: mask=0 | data_size=4B | dims/strides/tile dims
            su8 g1;
            g1[0] = 0x00020000u;                           // data_size=2 (4 bytes)
            g1[1] = ((uint32_t)IMG_W & 0xFFFFu) << 16;     // tensor_dim0 [15:0]
            g1[2] = ((uint32_t)IMG_W >> 16)                // tensor_dim0 [31:16]
                    | (((uint32_t)IMG_H & 0xFFFFu) << 16); // tensor_dim1 [15:0]
            g1[3] = ((uint32_t)IMG_H >> 16)                // tensor_dim1 [31:16]
                    | ((uint32_t)LDS_W << 16);             // tile_dim0 = 136
            g1[4] = (uint32_t)LDS_ROWS;                    // tile_dim1 = 18 (2D)
            g1[5] = (uint32_t)IMG_W;                       // tensor_dim0_stride lo
            g1[6] = 0u;                                    // stride0 hi | stride1 lo
            g1[7] = 0u;                                    // stride1 hi
            su4 g2 = {0u, 0u, 0u, 0u};                     // unused (2D tensor)
            su4 g3 = {0u, 0u, 0u, 0u};

            asm volatile("tensor_load_to_lds %0, %1, %2, %3"
                         :
                         : "s"(g0), "s"(g1), "s"(g2), "s"(g3)
                         : "memory");
            __builtin_amdgcn_s_wait_tensorcnt(0);
        }
    } else {
        // ---- Edge tiles: per-lane async global->LDS b128 with replicate clamp
        // (replicate padding == -inf padding for max-pool: the replicated value
        // is already inside the window).
        for (int i = tid; i < NLOADS; i += NTHREADS) {
            const int row = i / NVEC;                      // 0..17
            const int v   = i - row * NVEC;                // 0..33
            const int gr  = clampi(gy - 1 + row, 0, IMG_H - 1);
            const int gc  = clampi(base_x + v * 4, 0, IMG_W - 4);
            const uint64_t gaddr =
                (uint64_t)(uintptr_t)(rep + (size_t)gr * IMG_W + gc);
            const uint32_t laddr =
                (uint32_t)(uintptr_t)(&smem[row * LDS_P + v * 4]);
            asm volatile("global_load_async_to_lds_b128 %0, %1, off"
                         :
                         : "v"(laddr), "v"(gaddr)
                         : "memory");
        }
        __builtin_amdgcn_s_wait_asynccnt(0);
    }
    __syncthreads();                                       // tile is in LDS

    // LDS rows ty, ty+1, ty+2  <->  global rows py-1, py, py+1
    const int r0 = (int)threadIdx.y * LDS_P;

    // Vertical max over 3 rows for the 6 columns px-1 .. px+4
    float vmax[6], mid[6];
#pragma unroll
    for (int k = 0; k < 6; ++k) {
        const int gxc = clampi(px - 1 + k, 0, IMG_W - 1);  // replicate clamp
        const int li  = gxc - base_x;
        const float a = smem[r0 + li];
        const float b = smem[r0 + LDS_P + li];
        const float c = smem[r0 + 2 * LDS_P + li];
        mid[k]  = b;
        vmax[k] = fmaxf(fmaxf(a, b), c);
    }

    v4f o4;
#pragma unroll
    for (int j = 0; j < 4; ++j) {
        const float hm  = fmaxf(fmaxf(vmax[j], vmax[j + 1]), vmax[j + 2]);
        const float cen = mid[j + 1];                      // exact center value
        const bool  hit = (cen == hm) && (cen >= rep_thr) && (r4[j] >= rel_thr);
        o4[j] = hit ? 1.0f : 0.0f;
    }
    __builtin_nontemporal_store(o4, (v4f*)(out + (size_t)py * IMG_W + px));
}

extern "C" void kernel_launch(void* const* d_in, const int* in_sizes, int n_in,
                              void* d_out, int out_size, void* d_ws, size_t ws_size,
                              hipStream_t stream) {
    (void)in_sizes; (void)n_in; (void)out_size; (void)d_ws; (void)ws_size;
    const float* rel     = (const float*)d_in[0];
    const float* rep     = (const float*)d_in[1];
    const float* rel_thr = (const float*)d_in[2];
    const float* rep_thr = (const float*)d_in[3];
    float*       out     = (float*)d_out;

    dim3 block(32, 16);                         // 512 threads = 16 wave32s
    dim3 grid(IMG_W / TILE_X, IMG_H / TILE_Y);  // 32 x 256 tiles
    hipLaunchKernelGGL(nms_kernel, grid, block, 0, stream,
                       rel, rep, rel_thr, rep_thr, out);
}